// paramDLRM_Net_12790412607703
// MI455X (gfx1250) — compile-verified
//
#include <hip/hip_runtime.h>
#include <hip/hip_bf16.h>
#include <stddef.h>

// ---------------- problem constants (match reference) ----------------
#define RWS   8        // WORLD_SIZE
#define NF    8        // NUM_FEATURES
#define NB_   4096     // BATCH
#define RB    32768    // RWS*NB_
#define NBAGS 262144   // RWS*NF*NB_
#define EMB_ROWS 200000
#define EMB_DIM  64
#define MAXP  136      // 128 max bag + 8 zero-pad slack (keeps unconditional reads safe)
#define SCAN_BLK 256
#define NBLOCKS (NBAGS / SCAN_BLK)   // 1024
#define WPB 8                         // waves per block in pooling kernel

typedef __attribute__((ext_vector_type(2))) float v2f;
typedef __attribute__((ext_vector_type(4))) float v4f;
typedef __attribute__((ext_vector_type(8))) float v8f;

// input bag for output bag o=(f,r,b): in = r*F*B + f*B + b
__device__ __forceinline__ int perm_in(int o) {
    int f = o >> 15;          // / RB
    int j = o & (RB - 1);
    int r = j >> 12;          // / NB_
    int b = j & (NB_ - 1);
    return (r << 15) + (f << 12) + b;   // r*NF*NB_ + f*NB_ + b
}

// ---------------- K1: per-block exclusive scans (both orders) ----------------
__global__ void scan_block_kernel(const int* __restrict__ lengths,
                                  int* __restrict__ partialA, int* __restrict__ partialB,
                                  int* __restrict__ bsumA, int* __restrict__ bsumB) {
    __shared__ int sA[SCAN_BLK], sB[SCAN_BLK];
    const int tid = threadIdx.x;
    const int g = blockIdx.x * SCAN_BLK + tid;
    const int va = lengths[g];               // input order
    const int vb = lengths[perm_in(g)];      // permuted (f,r,b) order
    sA[tid] = va; sB[tid] = vb;
    __syncthreads();
    for (int d = 1; d < SCAN_BLK; d <<= 1) {
        int ta = (tid >= d) ? sA[tid - d] : 0;
        int tb = (tid >= d) ? sB[tid - d] : 0;
        __syncthreads();
        sA[tid] += ta; sB[tid] += tb;
        __syncthreads();
    }
    partialA[g] = sA[tid] - va;              // exclusive within block
    partialB[g] = sB[tid] - vb;
    if (tid == SCAN_BLK - 1) { bsumA[blockIdx.x] = sA[tid]; bsumB[blockIdx.x] = sB[tid]; }
}

// ---------------- K2: exclusive scan of the 1024 block sums ----------------
__global__ void scan_sums_kernel(const int* __restrict__ bsumA, const int* __restrict__ bsumB,
                                 int* __restrict__ bscanA, int* __restrict__ bscanB) {
    __shared__ int sA[NBLOCKS], sB[NBLOCKS];
    const int tid = threadIdx.x;
    const int va = bsumA[tid], vb = bsumB[tid];
    sA[tid] = va; sB[tid] = vb;
    __syncthreads();
    for (int d = 1; d < NBLOCKS; d <<= 1) {
        int ta = (tid >= d) ? sA[tid - d] : 0;
        int tb = (tid >= d) ? sB[tid - d] : 0;
        __syncthreads();
        sA[tid] += ta; sB[tid] += tb;
        __syncthreads();
    }
    bscanA[tid] = sA[tid] - va;
    bscanB[tid] = sB[tid] - vb;
}

// ---------------- K3: offsets output (per-feature rebased, as float) ----------------
__global__ void offsets_kernel(const int* __restrict__ partialB, const int* __restrict__ bscanB,
                               float* __restrict__ out) {
    const int o = blockIdx.x * SCAN_BLK + threadIdx.x;
    const int base = (o >> 15) << 15;                       // f * RB
    const int offo = partialB[o]    + bscanB[o >> 8];
    const int offb = partialB[base] + bscanB[base >> 8];
    out[o] = (float)(offo - offb);                          // < 2^24 -> exact in f32
}

// ---------------- K4: one wave per bag: regather + WMMA pooling ----------------
// Dim mapping: ly dim d = 4*n + s  (n = lane&15 column, s = slice 0..3).
// => each lane's B contribution per row is one contiguous v4f (global_load_b128),
//    lanes 0-15 cover a full 256B row, lanes 16-31 the matching row of the other K-pair (2 rows on).
__global__ void __launch_bounds__(WPB * 32)
bag_pool_kernel(const int* __restrict__ lengths, const int* __restrict__ indices,
                const float* __restrict__ emb,
                const int* __restrict__ partialA, const int* __restrict__ partialB,
                const int* __restrict__ bscanA,  const int* __restrict__ bscanB,
                float* __restrict__ permuted_out, float* __restrict__ ly_out) {
    __shared__ int lds_idx[WPB][MAXP];
    const int wave = threadIdx.x >> 5;
    const int lane = threadIdx.x & 31;
    const int o = blockIdx.x * WPB + wave;

    const int f = o >> 15;
    const int in_bag = perm_in(o);
    const int L     = lengths[in_bag];
    const int start = partialA[in_bag] + bscanA[in_bag >> 8];   // off_in[in_bag]
    const int outb  = partialB[o]      + bscanB[o >> 8];        // off_out[o]

    // Stage bag indices to LDS; emit permuted (exact: idx < 2^24) with NT stores.
    for (int t = lane; t < L; t += 32) {
        const int idx = __builtin_nontemporal_load(&indices[start + t]);
        lds_idx[wave][t] = idx;
        __builtin_nontemporal_store((float)idx, &permuted_out[outb + t]);
    }
    if (lane < 8) lds_idx[wave][L + lane] = 0;     // zero-pad: unconditional reads are safe
    asm volatile("s_wait_dscnt 0" ::: "memory");   // LDS staging visible wave-wide

    const float* table = emb + (size_t)f * (size_t)EMB_ROWS * EMB_DIM;
    const int khalf = (lane < 16) ? 0 : 2;   // A/B K-pair owned by this half (ISA 7.12.2)
    const int col = lane & 15;               // N column
    const int coff = col * 4;                // contiguous float4 per lane

    v8f acc0 = {}, acc1 = {}, acc2 = {}, acc3 = {};
    const int Lfull = L & ~3;

    if (Lfull > 0) {
        // software-pipelined K-pair indices: (g+khalf) even -> aligned ds_load_b64
        int2 ip = *reinterpret_cast<const int2*>(&lds_idx[wave][khalf]);
        const v2f ones = {1.0f, 1.0f};
        for (int g = 0; g < Lfull; g += 4) {
            const int2 ipn = *reinterpret_cast<const int2*>(&lds_idx[wave][g + 4 + khalf]);
            const float* p0 = table + (size_t)ip.x * EMB_DIM;
            const float* p1 = table + (size_t)ip.y * EMB_DIM;
            __builtin_prefetch(table + (size_t)ipn.x * EMB_DIM + coff, 0, 3);
            __builtin_prefetch(table + (size_t)ipn.y * EMB_DIM + coff, 0, 3);
            const v4f ra = *reinterpret_cast<const v4f*>(p0 + coff);
            const v4f rb = *reinterpret_cast<const v4f*>(p1 + coff);
            const v2f b0 = {ra.x, rb.x}, b1 = {ra.y, rb.y}, b2 = {ra.z, rb.z}, b3 = {ra.w, rb.w};
            acc0 = __builtin_amdgcn_wmma_f32_16x16x4_f32(false, ones, false, b0, (short)0, acc0, false, false);
            acc1 = __builtin_amdgcn_wmma_f32_16x16x4_f32(false, ones, false, b1, (short)0, acc1, false, false);
            acc2 = __builtin_amdgcn_wmma_f32_16x16x4_f32(false, ones, false, b2, (short)0, acc2, false, false);
            acc3 = __builtin_amdgcn_wmma_f32_16x16x4_f32(false, ones, false, b3, (short)0, acc3, false, false);
            ip = ipn;
        }
    }
    if (Lfull < L) {                        // masked tail group (1..3 valid rows)
        const int rem = L - Lfull;
        v2f a; a.x = (khalf < rem) ? 1.0f : 0.0f;
               a.y = (khalf + 1 < rem) ? 1.0f : 0.0f;
        const int2 ip = *reinterpret_cast<const int2*>(&lds_idx[wave][Lfull + khalf]); // zero-padded
        const float* p0 = table + (size_t)ip.x * EMB_DIM;   // row 0 when padded: masked by A
        const float* p1 = table + (size_t)ip.y * EMB_DIM;
        const v4f ra = *reinterpret_cast<const v4f*>(p0 + coff);
        const v4f rb = *reinterpret_cast<const v4f*>(p1 + coff);
        const v2f b0 = {ra.x, rb.x}, b1 = {ra.y, rb.y}, b2 = {ra.z, rb.z}, b3 = {ra.w, rb.w};
        acc0 = __builtin_amdgcn_wmma_f32_16x16x4_f32(false, a, false, b0, (short)0, acc0, false, false);
        acc1 = __builtin_amdgcn_wmma_f32_16x16x4_f32(false, a, false, b1, (short)0, acc1, false, false);
        acc2 = __builtin_amdgcn_wmma_f32_16x16x4_f32(false, a, false, b2, (short)0, acc2, false, false);
        acc3 = __builtin_amdgcn_wmma_f32_16x16x4_f32(false, a, false, b3, (short)0, acc3, false, false);
    }

    // D row M=0 (lanes 0-15, VGPR0 of C/D): slices s=0..3 are dims 4*col+0..3 -> one b128 NT store.
    if (lane < 16) {
        v4f r; r.x = acc0[0]; r.y = acc1[0]; r.z = acc2[0]; r.w = acc3[0];
        float* lyp = ly_out + (size_t)o * EMB_DIM + coff;
        __builtin_nontemporal_store(r, reinterpret_cast<v4f*>(lyp));
    }
}

// ---------------- host ----------------
extern "C" void kernel_launch(void* const* d_in, const int* in_sizes, int n_in,
                              void* d_out, int out_size, void* d_ws, size_t ws_size,
                              hipStream_t stream) {
    const int*   lengths = (const int*)d_in[0];
    const int*   indices = (const int*)d_in[1];
    const float* emb     = (const float*)d_in[2];
    const int n = in_sizes[0];           // NBAGS = 262144
    const int T = in_sizes[1];           // total index count

    int* ws = (int*)d_ws;                // 2*n + 4*NBLOCKS ints ≈ 2.1 MB
    int* partialA = ws;
    int* partialB = partialA + n;
    int* bsumA  = partialB + n;
    int* bsumB  = bsumA + NBLOCKS;
    int* bscanA = bsumB + NBLOCKS;
    int* bscanB = bscanA + NBLOCKS;

    float* out_offsets  = (float*)d_out;          // [n]
    float* out_permuted = out_offsets + n;        // [T]
    float* out_ly       = out_permuted + T;       // [n * 64]

    scan_block_kernel<<<NBLOCKS, SCAN_BLK, 0, stream>>>(lengths, partialA, partialB, bsumA, bsumB);
    scan_sums_kernel<<<1, NBLOCKS, 0, stream>>>(bsumA, bsumB, bscanA, bscanB);
    offsets_kernel<<<NBLOCKS, SCAN_BLK, 0, stream>>>(partialB, bscanB, out_offsets);
    bag_pool_kernel<<<n / WPB, WPB * 32, 0, stream>>>(lengths, indices, emb,
                                                      partialA, partialB, bscanA, bscanB,
                                                      out_permuted, out_ly);
}